// GATN_72756745994560
// MI455X (gfx1250) — compile-verified
//
#include <hip/hip_runtime.h>
#include <hip/hip_bf16.h>

// Problem constants (match reference)
static constexpr int N_NODES = 50000;
static constexpr int N_EDGES = 800000;
static constexpr int D       = 128;   // F_IN == HC
static constexpr int H       = 4;
static constexpr int C       = 32;

typedef __attribute__((ext_vector_type(2))) float v2f;
typedef __attribute__((ext_vector_type(8))) float v8f;

// ---------- ordered-float encoding for atomicMax on signed floats ----------
__device__ __forceinline__ unsigned enc_ord(float x) {
    unsigned u = __float_as_uint(x);
    return (u & 0x80000000u) ? ~u : (u | 0x80000000u);
}
__device__ __forceinline__ float dec_ord(unsigned u) {
    return __uint_as_float((u & 0x80000000u) ? (u & 0x7FFFFFFFu) : ~u);
}
static constexpr unsigned ENC_NEG_INF = 0x007FFFFFu; // enc(-inf)

// ---------------------- GraphNorm statistics -------------------------------
__global__ void zero_stats_kernel(float* stats) {
    int t = threadIdx.x;
    if (t < 256) stats[t] = 0.f;   // sum[128] + sumsq[128]
}

__global__ __launch_bounds__(128) void col_stats_kernel(
    const float* __restrict__ h, float* __restrict__ sum, float* __restrict__ sumsq, int nrows)
{
    int col = threadIdx.x;                 // 128 columns
    float s = 0.f, s2 = 0.f;
    for (int r = blockIdx.x; r < nrows; r += gridDim.x) {
        float v = h[(size_t)r * D + col];
        s += v; s2 += v * v;
    }
    atomicAdd(&sum[col], s);
    atomicAdd(&sumsq[col], s2);
}

// stats layout: [0:128)=sum [128:256)=sumsq [256:384)=scaleA [384:512)=shiftB
__global__ void finalize_stats_kernel(
    float* __restrict__ stats, const float* __restrict__ gw,
    const float* __restrict__ gb, const float* __restrict__ gms, int nrows)
{
    int c = threadIdx.x;
    if (c >= D) return;
    float invN  = 1.0f / (float)nrows;
    float mean  = stats[c] * invN;
    float msq   = stats[128 + c] * invN;
    float ms    = gms[c];
    // var = E[(h - ms*mean)^2] = E[h^2] - mean^2 * ms * (2 - ms)
    float var   = msq - mean * mean * ms * (2.0f - ms);
    float inv   = rsqrtf(var + 1e-5f);
    float a     = gw[c] * inv;
    stats[256 + c] = a;                       // scale
    stats[384 + c] = gb[c] - a * ms * mean;   // shift
}

// ------------- fused GraphNorm + dual GEMM (xl = hn@Wl, xr = hn@Wr) --------
// Uses V_WMMA_F32_16X16X4_F32. Block = 256 thr (8 waves) handles 32 rows.
// wave w: row-tile rt = w>>2, col quadrant q = w&3 -> col tiles {2q,2q+1}
// for BOTH Wl and Wr (4 accumulators of v8f).
__global__ __launch_bounds__(256) void gemm_norm_xlxr_kernel(
    const float* __restrict__ hin, const float* __restrict__ stats,
    const float* __restrict__ Wl,  const float* __restrict__ Wr,
    float* __restrict__ xl, float* __restrict__ xr, int nrows)
{
    __shared__ float lds[32 * 132];           // padded stride 132 -> no bank conflicts
    const float* scaleA = stats + 256;
    const float* shiftB = stats + 384;

    int tid = threadIdx.x;
    int rowbase = blockIdx.x * 32;

    // stage normalized 32x128 tile
    for (int i = tid; i < 32 * D; i += 256) {
        int r = i >> 7, c = i & 127;
        int gr = rowbase + r;
        float v = 0.f;
        if (gr < nrows) v = hin[(size_t)gr * D + c] * scaleA[c] + shiftB[c];
        lds[r * 132 + c] = v;
    }
    __syncthreads();

    int wave = tid >> 5, lane = tid & 31;
    int rt   = wave >> 2;
    int q    = wave & 3;
    int m    = lane & 15;
    int khalf = (lane >> 4) << 1;             // 0 or 2
    int n0 = (2 * q) * 16 + (lane & 15);
    int n1 = n0 + 16;

    v8f accL0 = {}, accL1 = {}, accR0 = {}, accR1 = {};
    const float* ldsA = &lds[(rt * 16 + m) * 132];

    #pragma unroll 4
    for (int k0 = 0; k0 < D; k0 += 4) {
        int kb = k0 + khalf;
        v2f a = *(const v2f*)(ldsA + kb);     // A[m][kb], A[m][kb+1]
        v2f bL0, bL1, bR0, bR1;
        bL0.x = Wl[kb * D + n0]; bL0.y = Wl[(kb + 1) * D + n0];
        bL1.x = Wl[kb * D + n1]; bL1.y = Wl[(kb + 1) * D + n1];
        bR0.x = Wr[kb * D + n0]; bR0.y = Wr[(kb + 1) * D + n0];
        bR1.x = Wr[kb * D + n1]; bR1.y = Wr[(kb + 1) * D + n1];
        accL0 = __builtin_amdgcn_wmma_f32_16x16x4_f32(false, a, false, bL0, (short)0, accL0, false, false);
        accL1 = __builtin_amdgcn_wmma_f32_16x16x4_f32(false, a, false, bL1, (short)0, accL1, false, false);
        accR0 = __builtin_amdgcn_wmma_f32_16x16x4_f32(false, a, false, bR0, (short)0, accR0, false, false);
        accR1 = __builtin_amdgcn_wmma_f32_16x16x4_f32(false, a, false, bR1, (short)0, accR1, false, false);
    }

    int col0 = 2 * q * 16 + (lane & 15);
    #pragma unroll
    for (int i = 0; i < 8; ++i) {
        int row = rowbase + rt * 16 + i + ((lane >> 4) << 3);
        if (row < nrows) {
            size_t base = (size_t)row * D;
            xl[base + col0]      = accL0[i];
            xl[base + col0 + 16] = accL1[i];
            xr[base + col0]      = accR0[i];
            xr[base + col0 + 16] = accR1[i];
        }
    }
}

// ---------------------- per-layer re-initialization ------------------------
__global__ void init_md_kernel(unsigned* __restrict__ m_enc, float* __restrict__ denom, int n) {
    int i = blockIdx.x * blockDim.x + threadIdx.x;
    if (i < n) { m_enc[i] = ENC_NEG_INF; denom[i] = 0.f; }
}

__global__ void init_out_kernel(float* __restrict__ hout, const float* __restrict__ bias, int n) {
    int i = blockIdx.x * blockDim.x + threadIdx.x;
    if (i < n) hout[i] = bias[i & (D - 1)];
}

// ---------------- edge pass 1: logits + segment max (wave per edge) --------
// Feature layout per lane: f = lane*4 + q  (float4/b128 gathers); all 4
// elements of a lane belong to head j = lane>>3, so the per-head dot needs a
// per-lane FMA chain + 3-step shfl_xor reduction over groups of 8 lanes.
__global__ __launch_bounds__(256) void edge_logits_max_kernel(
    const float* __restrict__ xl, const float* __restrict__ xr,
    const int* __restrict__ src, const int* __restrict__ dst,
    const float* __restrict__ att, float* __restrict__ logits,
    unsigned* __restrict__ m_enc, int nedges)
{
    int e = blockIdx.x * 8 + (threadIdx.x >> 5);
    if (e >= nedges) return;
    int lane = threadIdx.x & 31;
    int s = src[e], d = dst[e];

    float4 a = ((const float4*)(xl + (size_t)s * D))[lane];
    float4 b = ((const float4*)(xr + (size_t)d * D))[lane];
    float4 t = ((const float4*)att)[lane];

    float v0 = a.x + b.x; v0 = (v0 > 0.f) ? v0 : 0.2f * v0;
    float v1 = a.y + b.y; v1 = (v1 > 0.f) ? v1 : 0.2f * v1;
    float v2 = a.z + b.z; v2 = (v2 > 0.f) ? v2 : 0.2f * v2;
    float v3 = a.w + b.w; v3 = (v3 > 0.f) ? v3 : 0.2f * v3;
    float p = v0 * t.x + v1 * t.y + v2 * t.z + v3 * t.w;

    p += __shfl_xor(p, 1, 32);
    p += __shfl_xor(p, 2, 32);
    p += __shfl_xor(p, 4, 32);                 // per-head sum in every lane of 8-group

    if ((lane & 7) == 0) {
        int j = lane >> 3;                      // head index
        logits[(size_t)e * H + j] = p;
        atomicMax(&m_enc[(size_t)d * H + j], enc_ord(p));
    }
}

// ---------------- edge pass 2: a = exp(logit - m[dst]); denom += a ---------
__global__ __launch_bounds__(256) void edge_exp_denom_kernel(
    float* __restrict__ logits, const unsigned* __restrict__ m_enc,
    const int* __restrict__ dst, float* __restrict__ denom, int nedges)
{
    int e = blockIdx.x * blockDim.x + threadIdx.x;
    if (e >= nedges) return;
    int d = dst[e];
    float4 lg = ((const float4*)logits)[e];
    uint4  mu = ((const uint4*)m_enc)[d];
    float4 a;
    a.x = __expf(lg.x - dec_ord(mu.x));
    a.y = __expf(lg.y - dec_ord(mu.y));
    a.z = __expf(lg.z - dec_ord(mu.z));
    a.w = __expf(lg.w - dec_ord(mu.w));
    ((float4*)logits)[e] = a;                  // overwrite logits with a
    float* dn = denom + (size_t)d * H;
    atomicAdd(&dn[0], a.x);
    atomicAdd(&dn[1], a.y);
    atomicAdd(&dn[2], a.z);
    atomicAdd(&dn[3], a.w);
}

// ---------------- edge pass 3: scatter messages (wave per edge) ------------
// b128 gather of xl[src], per-lane head coefficient via dynamic shfl, then 4
// f32 atomic-adds at the gathered addresses (128 dword-atomics per edge).
__global__ __launch_bounds__(256) void edge_scatter_kernel(
    const float* __restrict__ xl, const float* __restrict__ a_buf,
    const float* __restrict__ denom, const int* __restrict__ src,
    const int* __restrict__ dst, const float* __restrict__ ew,
    float* __restrict__ hout, int nedges)
{
    int e = blockIdx.x * 8 + (threadIdx.x >> 5);
    if (e >= nedges) return;
    int lane = threadIdx.x & 31;
    int s = src[e], d = dst[e];
    float w = ew[e];

    float cv = 0.f;
    if (lane < H)
        cv = a_buf[(size_t)e * H + lane] / (denom[(size_t)d * H + lane] + 1e-16f) * w;
    float c = __shfl(cv, lane >> 3, 32);       // coefficient of this lane's head

    float4 v = ((const float4*)(xl + (size_t)s * D))[lane];
    float* op = hout + (size_t)d * D + lane * 4;
    atomicAdd(&op[0], v.x * c);
    atomicAdd(&op[1], v.y * c);
    atomicAdd(&op[2], v.z * c);
    atomicAdd(&op[3], v.w * c);
}

// ---------------- leaky_relu(0.01) between layers --------------------------
__global__ void lrelu_kernel(float* __restrict__ h, int n) {
    int i = blockIdx.x * blockDim.x + threadIdx.x;
    if (i < n) {
        float v = h[i];
        h[i] = (v > 0.f) ? v : 0.01f * v;
    }
}

// ---------------------------------------------------------------------------
extern "C" void kernel_launch(void* const* d_in, const int* in_sizes, int n_in,
                              void* d_out, int out_size, void* d_ws, size_t ws_size,
                              hipStream_t stream)
{
    const float* x  = (const float*)d_in[0];
    const int*   ei = (const int*)d_in[1];
    const float* ew = (const float*)d_in[2];
    const int* src = ei;
    const int* dst = ei + N_EDGES;

    // Workspace carve-up (~92 MB)
    float*    ws     = (float*)d_ws;
    float*    xl     = ws;
    float*    xr     = xl + (size_t)N_NODES * D;
    float*    logits = xr + (size_t)N_NODES * D;                  // E*H, reused as "a"
    unsigned* m_enc  = (unsigned*)(logits + (size_t)N_EDGES * H); // N*H
    float*    denom  = (float*)m_enc + (size_t)N_NODES * H;       // N*H
    float*    hbuf   = denom + (size_t)N_NODES * H;               // N*D
    float*    stats  = hbuf + (size_t)N_NODES * D;                // 512 floats

    for (int l = 0; l < 3; ++l) {
        const float* gw   = (const float*)d_in[3 + 7 * l + 0];
        const float* gb   = (const float*)d_in[3 + 7 * l + 1];
        const float* gms  = (const float*)d_in[3 + 7 * l + 2];
        const float* Wl   = (const float*)d_in[3 + 7 * l + 3];
        const float* Wr   = (const float*)d_in[3 + 7 * l + 4];
        const float* att  = (const float*)d_in[3 + 7 * l + 5];
        const float* bias = (const float*)d_in[3 + 7 * l + 6];

        const float* hin = (l == 0) ? x : hbuf;
        float* hout = (l == 2) ? (float*)d_out : hbuf;

        // GraphNorm stats -> per-column affine
        zero_stats_kernel<<<1, 256, 0, stream>>>(stats);
        col_stats_kernel<<<512, 128, 0, stream>>>(hin, stats, stats + 128, N_NODES);
        finalize_stats_kernel<<<1, 128, 0, stream>>>(stats, gw, gb, gms, N_NODES);

        // Fused norm + dual GEMM via v_wmma_f32_16x16x4_f32
        gemm_norm_xlxr_kernel<<<(N_NODES + 31) / 32, 256, 0, stream>>>(
            hin, stats, Wl, Wr, xl, xr, N_NODES);

        // Reset per-layer accumulators (stream order guarantees GEMM done; hin==hbuf safe)
        init_md_kernel<<<(N_NODES * H + 255) / 256, 256, 0, stream>>>(m_enc, denom, N_NODES * H);
        init_out_kernel<<<(N_NODES * D + 255) / 256, 256, 0, stream>>>(hout, bias, N_NODES * D);

        // Edge phase
        edge_logits_max_kernel<<<(N_EDGES + 7) / 8, 256, 0, stream>>>(
            xl, xr, src, dst, att, logits, m_enc, N_EDGES);
        edge_exp_denom_kernel<<<(N_EDGES + 255) / 256, 256, 0, stream>>>(
            logits, m_enc, dst, denom, N_EDGES);
        edge_scatter_kernel<<<(N_EDGES + 7) / 8, 256, 0, stream>>>(
            xl, logits, denom, src, dst, ew, hout, N_EDGES);

        if (l < 2)
            lrelu_kernel<<<(N_NODES * D + 255) / 256, 256, 0, stream>>>(hbuf, N_NODES * D);
    }
}